// TopKListwiseLoss_88175678587657
// MI455X (gfx1250) — compile-verified
//
#include <hip/hip_runtime.h>
#include <hip/hip_bf16.h>
#include <limits.h>

#define BD      32768      // docs per row (fixed by reference)
#define KTOP    10
#define NTHREADS 256
#define NWAVES  (NTHREADS / 32)
#define TILE    1024       // floats per LDS tile = NTHREADS * 4
#define NTILES  (BD / TILE)
#define NBUF    4          // async pipeline depth (prefetch distance 3)

// ---- CDNA5 async global->LDS path (guarded; fallback = direct b128 loads) ----
#if defined(__has_builtin)
#  if __has_builtin(__builtin_amdgcn_global_load_async_to_lds_b128) && \
      __has_builtin(__builtin_amdgcn_s_wait_asynccnt)
#    define USE_ASYNC 1
#  else
#    define USE_ASYNC 0
#  endif
#else
#  define USE_ASYNC 0
#endif

#if USE_ASYNC
typedef int v4i __attribute__((vector_size(16)));
typedef __attribute__((address_space(1))) v4i* g1_v4i_p;
typedef __attribute__((address_space(3))) v4i* l3_v4i_p;
// cpol = 1 -> TH_NT: single-use stream, don't retain in cache hierarchy
#define ASYNC_LOAD_B128(gptr, lptr)                                           \
  __builtin_amdgcn_global_load_async_to_lds_b128(                             \
      (g1_v4i_p)(gptr), (l3_v4i_p)(lptr), 0, 1)
#define ASYNC_WAIT(n) __builtin_amdgcn_s_wait_asynccnt(n)
#endif

__device__ __forceinline__ void topk_insert(float (&tv)[KTOP], int (&ti)[KTOP],
                                            float v, int idx) {
  // strict '>' keeps lower-index-first among ties (lane scans indices ascending)
  if (v > tv[KTOP - 1]) {
    tv[KTOP - 1] = v; ti[KTOP - 1] = idx;
#pragma unroll
    for (int j = KTOP - 1; j > 0; --j) {
      if (tv[j] > tv[j - 1]) {
        float fv = tv[j]; tv[j] = tv[j - 1]; tv[j - 1] = fv;
        int   fi = ti[j]; ti[j] = ti[j - 1]; ti[j - 1] = fi;
      }
    }
  }
}

__global__ __launch_bounds__(NTHREADS)
void TopKNDCG_rows_kernel(const float* __restrict__ scores,
                          const unsigned char* __restrict__ mask,
                          float* __restrict__ row_loss) {
  const int row  = blockIdx.x;
  const int tid  = threadIdx.x;
  const int lane = tid & 31;
  const int wave = tid >> 5;

  const float*         __restrict__ srow = scores + (size_t)row * BD;
  const unsigned char* __restrict__ mrow = mask   + (size_t)row * BD;

  const float NEGINF = -__builtin_inff();
  float tv[KTOP]; int ti[KTOP];
#pragma unroll
  for (int j = 0; j < KTOP; ++j) { tv[j] = NEGINF; ti[j] = INT_MAX; }

  // ---------------- streaming phase: per-lane register top-10 ----------------
#if USE_ASYNC
  __shared__ float sbuf[NBUF][TILE];
  // Each thread async-copies its own 16B per tile and later reads back exactly
  // those bytes -> only its own ASYNCcnt matters, no barriers needed.
  // Prologue: fill the pipeline with tiles 0..2 (prefetch distance NBUF-1 = 3).
#pragma unroll
  for (int t = 0; t < NBUF - 1; ++t)
    ASYNC_LOAD_B128(srow + t * TILE + tid * 4, &sbuf[t][tid * 4]);

#pragma unroll 4
  for (int t = 0; t < NTILES; ++t) {
    const int cur = t & (NBUF - 1);
    if (t + NBUF - 1 < NTILES) {
      ASYNC_LOAD_B128(srow + (t + NBUF - 1) * TILE + tid * 4,
                      &sbuf[(t + NBUF - 1) & (NBUF - 1)][tid * 4]);
      ASYNC_WAIT(NBUF - 1);          // tile t complete, 3 still in flight
    } else if (t + NBUF - 1 == NTILES) {
      ASYNC_WAIT(2);
    } else if (t + NBUF - 2 == NTILES) {
      ASYNC_WAIT(1);
    } else {
      ASYNC_WAIT(0);
    }
    const float4 v = *(const float4*)(&sbuf[cur][tid * 4]);
    const int base = t * TILE + tid * 4;
    topk_insert(tv, ti, v.x, base + 0);
    topk_insert(tv, ti, v.y, base + 1);
    topk_insert(tv, ti, v.z, base + 2);
    topk_insert(tv, ti, v.w, base + 3);
  }
#else
#pragma unroll 4
  for (int t = 0; t < NTILES; ++t) {
    const float4 v = ((const float4*)srow)[t * NTHREADS + tid];
    const int base = (t * NTHREADS + tid) * 4;
    topk_insert(tv, ti, v.x, base + 0);
    topk_insert(tv, ti, v.y, base + 1);
    topk_insert(tv, ti, v.z, base + 2);
    topk_insert(tv, ti, v.w, base + 3);
  }
#endif

  // ---------------- wave merge: 32 sorted lists -> wave top-10 in LDS --------
  __shared__ float wl_v[NWAVES * KTOP];
  __shared__ int   wl_i[NWAVES * KTOP];

#pragma unroll
  for (int r = 0; r < KTOP; ++r) {
    float cv = tv[0]; int ci = ti[0];
#pragma unroll
    for (int s = 16; s > 0; s >>= 1) {
      const float ov = __shfl_xor(cv, s, 32);
      const int   oi = __shfl_xor(ci, s, 32);
      if (ov > cv || (ov == cv && oi < ci)) { cv = ov; ci = oi; }
    }
    if (ti[0] == ci) {  // winner lane pops its head (doc indices are unique)
#pragma unroll
      for (int j = 0; j < KTOP - 1; ++j) { tv[j] = tv[j + 1]; ti[j] = ti[j + 1]; }
      tv[KTOP - 1] = NEGINF; ti[KTOP - 1] = INT_MAX;
    }
    if (lane == 0) { wl_v[wave * KTOP + r] = cv; wl_i[wave * KTOP + r] = ci; }
  }
  __syncthreads();

  // ---------------- wave 0: merge 8 sorted lists, compute NDCG loss ----------
  if (tid < 32) {
    int p = 0;
    float mylab = 0.0f;
#pragma unroll
    for (int r = 0; r < KTOP; ++r) {
      float cv0; int ci0;
      if (lane < NWAVES && p < KTOP) { cv0 = wl_v[lane * KTOP + p]; ci0 = wl_i[lane * KTOP + p]; }
      else                           { cv0 = NEGINF;                ci0 = INT_MAX; }
      float cv = cv0; int ci = ci0;
#pragma unroll
      for (int s = 16; s > 0; s >>= 1) {
        const float ov = __shfl_xor(cv, s, 32);
        const int   oi = __shfl_xor(ci, s, 32);
        if (ov > cv || (ov == cv && oi < ci)) { cv = ov; ci = oi; }
      }
      if (ci0 == ci && lane < NWAVES && p < KTOP) ++p;   // winner list advances
      if (lane == r) mylab = (ci != INT_MAX) ? (float)mrow[ci] : 0.0f;
    }
    // discount for position lane+1 : 1/log2(lane+2)
    const float invd = (lane < KTOP) ? (1.0f / log2f((float)(lane + 2))) : 0.0f;
    float a = (mylab > 0.0f) ? invd : 0.0f;                 // actual DCG term
    const unsigned bal = (unsigned)__ballot(mylab > 0.0f);
    const int npos = __builtin_popcount(bal);
    float b = (lane < npos) ? invd : 0.0f;                  // ideal DCG term
#pragma unroll
    for (int s = 16; s > 0; s >>= 1) {
      a += __shfl_xor(a, s, 32);
      b += __shfl_xor(b, s, 32);
    }
    if (lane == 0) row_loss[row] = (b > 0.0f) ? (1.0f - a / b) : 0.0f;
  }
}

// Deterministic fixed-order final reduction (single block).
__global__ __launch_bounds__(NTHREADS)
void TopKNDCG_reduce_kernel(const float* __restrict__ row_loss,
                            float* __restrict__ out, int n) {
  __shared__ float s[NTHREADS];
  float acc = 0.0f;
  for (int i = threadIdx.x; i < n; i += NTHREADS) acc += row_loss[i];
  s[threadIdx.x] = acc;
  __syncthreads();
  for (int step = NTHREADS / 2; step > 0; step >>= 1) {
    if (threadIdx.x < step) s[threadIdx.x] += s[threadIdx.x + step];
    __syncthreads();
  }
  if (threadIdx.x == 0) out[0] = s[0] / (float)n;
}

extern "C" void kernel_launch(void* const* d_in, const int* in_sizes, int n_in,
                              void* d_out, int out_size, void* d_ws, size_t ws_size,
                              hipStream_t stream) {
  const float*         scores = (const float*)d_in[0];
  const unsigned char* mask   = (const unsigned char*)d_in[1];  // jnp.bool_ = 1 byte/elem
  float* out      = (float*)d_out;
  float* row_loss = (float*)d_ws;   // B_Q floats of scratch

  const int bq = in_sizes[0] / BD;  // 4096 for the reference shapes

  TopKNDCG_rows_kernel<<<bq, NTHREADS, 0, stream>>>(scores, mask, row_loss);
  TopKNDCG_reduce_kernel<<<1, NTHREADS, 0, stream>>>(row_loss, out, bq);
}